// TopkRouter_85478439125272
// MI455X (gfx1250) — compile-verified
//
#include <hip/hip_runtime.h>
#include <hip/hip_bf16.h>
#include <math.h>
#include <stdint.h>

// ---------------------------------------------------------------------------
// MoE router (DeepSeek-style) for MI455X / gfx1250.
//   logits = hidden[32768,4096] @ weight[256,4096]^T  (fp32 V_WMMA_F32_16X16X4_F32)
//   scores = sigmoid(logits) -> grouped top-k routing
// HBM-bound: 512MB hidden @ 23.3 TB/s ~= 23us floor, need only ~3 PFLOP/s fp32
// -> keep full fp32 parity. Staging now uses the Tensor Data Mover (TDM) with
// double-buffered LDS so DMA overlaps WMMA; TDM's pad feature inserts the
// bank-conflict padding (2 DWORDs per 16) during the DMA itself.
// ---------------------------------------------------------------------------

typedef __attribute__((ext_vector_type(2))) float v2f;
typedef __attribute__((ext_vector_type(8))) float v8f;
typedef __attribute__((ext_vector_type(4))) unsigned int u32x4;
typedef __attribute__((ext_vector_type(4))) int i32x4;
typedef __attribute__((ext_vector_type(8))) int i32x8;

constexpr int kTokens   = 32768;
constexpr int kDim      = 4096;
constexpr int kExperts  = 256;
constexpr int kTopK     = 8;
constexpr int kGroups   = 8;
constexpr int kGroupSz  = 32;
constexpr int kTopGroup = 4;
constexpr float kScale  = 2.5f;

// GEMM tiling
constexpr int BT   = 128;  // tokens per workgroup (8 M-tiles of 16)
constexpr int KC   = 16;   // K-slab per buffer
constexpr int LDSP = 18;   // 16 data + 2 pad DWORDs: 8B-aligned, conflict-free

#if __has_builtin(__builtin_amdgcn_tensor_load_to_lds)
#define USE_TDM 1
#else
#define USE_TDM 0
#endif

#if USE_TDM
// Issue one TDM 2D tile load: tile_d0 x tile_d1 elements (fp32), row stride
// `stride0` elements, destination LDS rows padded 16->18 DWORDs by the DMA.
__device__ __forceinline__ void tdm_load_2d(unsigned lds_addr, const float* gptr,
                                            unsigned tile_d0, unsigned tile_d1,
                                            unsigned tensor_d0, unsigned tensor_d1,
                                            unsigned stride0) {
  const unsigned long long ga = (unsigned long long)(size_t)gptr;
  u32x4 g0;
  g0[0] = 1u;                                   // count=1, user descriptor
  g0[1] = lds_addr;                             // LDS byte address
  g0[2] = (unsigned)(ga & 0xFFFFFFFFull);       // global_addr[31:0]
  g0[3] = (unsigned)((ga >> 32) & 0x01FFFFFFull) | (2u << 30);  // [56:32], type=2
  i32x8 g1;
  g1[0] = (int)((2u << 16)      // data_size = 4B
              | (1u << 20)      // pad_enable
              | (3u << 22)      // pad_interval: 16 DWORDs
              | (1u << 25));    // pad_amount:   2 DWORDs
  g1[1] = (int)((tensor_d0 & 0xFFFFu) << 16);                        // dim0[15:0]
  g1[2] = (int)(((tensor_d0 >> 16) & 0xFFFFu) | ((tensor_d1 & 0xFFFFu) << 16));
  g1[3] = (int)(((tensor_d1 >> 16) & 0xFFFFu) | ((tile_d0 & 0xFFFFu) << 16));
  g1[4] = (int)(tile_d1 & 0xFFFFu);                                  // tile_dim1
  g1[5] = (int)stride0;                                              // dim0_stride
  g1[6] = 0;
  g1[7] = 0;
  const i32x4 z4 = {0, 0, 0, 0};
#if defined(__clang_major__) && __clang_major__ >= 23
  const i32x8 z8 = {0, 0, 0, 0, 0, 0, 0, 0};
  __builtin_amdgcn_tensor_load_to_lds(g0, g1, z4, z4, z8, 0);
#else
  __builtin_amdgcn_tensor_load_to_lds(g0, g1, z4, z4, 0);
#endif
}
#endif  // USE_TDM

// ---------------------------------------------------------------------------
// Kernel 1: router GEMM + sigmoid -> scores in d_ws [32768 x 256] fp32.
// 8 waves/WG; wave w owns experts [32w,32w+32); WG shares 128 tokens.
// Double-buffered LDS slabs; TDM prefetches slab c+1 while slab c computes.
// ---------------------------------------------------------------------------
__launch_bounds__(256, 1)
__global__ void moe_router_gemm(const float* __restrict__ hidden,
                                const float* __restrict__ weight,
                                float* __restrict__ scores) {
  __shared__ float As[2][BT * LDSP];        // 2 x 128 x 18 (~18 KB)
  __shared__ float Ws[2][kExperts * LDSP];  // 2 x 256 x 18 (~37 KB)

  const int tid  = threadIdx.x;
  const int wave = tid >> 5;
  const int lane = tid & 31;
  const int l15  = lane & 15;
  const int h2   = (lane >> 4) << 1;   // K phase: 0 (lanes 0-15) / 2 (16-31)
  const int tok0 = blockIdx.x * BT;

  v8f acc[8][2];
#pragma unroll
  for (int mt = 0; mt < 8; ++mt)
#pragma unroll
    for (int nt = 0; nt < 2; ++nt) acc[mt][nt] = (v8f){0, 0, 0, 0, 0, 0, 0, 0};

  constexpr int nChunks = kDim / KC;  // 256

#if USE_TDM
  if (wave == 0) {  // wave-uniform branch; TDM is a wave-level op
    tdm_load_2d((unsigned)(size_t)&As[0][0], hidden + (size_t)tok0 * kDim,
                KC, BT, kDim, kTokens, kDim);
    tdm_load_2d((unsigned)(size_t)&Ws[0][0], weight,
                KC, kExperts, kDim, kExperts, kDim);
  }
#endif

  for (int c = 0; c < nChunks; ++c) {
    const int buf = c & 1;
#if USE_TDM
    if (wave == 0) {
      if (c + 1 < nChunks) {
        const int kb = (c + 1) * KC;
        tdm_load_2d((unsigned)(size_t)&As[buf ^ 1][0],
                    hidden + (size_t)tok0 * kDim + kb, KC, BT, kDim, kTokens, kDim);
        tdm_load_2d((unsigned)(size_t)&Ws[buf ^ 1][0],
                    weight + kb, KC, kExperts, kDim, kExperts, kDim);
        __builtin_amdgcn_s_wait_tensorcnt((short)2);  // chunk c done; c+1 in flight
      } else {
        __builtin_amdgcn_s_wait_tensorcnt((short)0);
      }
    }
    __syncthreads();  // release all waves to consume buf
#else
    // Fallback: cooperative staging (coalesced 64B rows, padded LDS rows)
    {
      const int kA = tid & 15;        // 16 k per row
      const int rA = tid >> 4;        // 16 rows per iteration
      const int kb = c * KC;
      __syncthreads();
#pragma unroll
      for (int r = 0; r < BT; r += 16)
        As[buf][(rA + r) * LDSP + kA] =
            hidden[(size_t)(tok0 + rA + r) * kDim + kb + kA];
#pragma unroll
      for (int r = 0; r < kExperts; r += 16)
        Ws[buf][(rA + r) * LDSP + kA] =
            weight[(size_t)(rA + r) * kDim + kb + kA];
      __syncthreads();
    }
#endif

#pragma unroll
    for (int ks = 0; ks < KC; ks += 4) {
      // B fragment (4x16): lane holds B[ks + h2 + v][n = l15], v = 0..1
      const float* wb = &Ws[buf][(wave * 32 + l15) * LDSP + ks + h2];
      v2f b0 = {wb[0], wb[1]};
      v2f b1 = {wb[16 * LDSP], wb[16 * LDSP + 1]};
#pragma unroll
      for (int mt = 0; mt < 8; ++mt) {
        // A fragment (16x4): lane holds A[m = l15][ks + h2 + v]
        const float* ab = &As[buf][(mt * 16 + l15) * LDSP + ks + h2];
        v2f a = {ab[0], ab[1]};
        acc[mt][0] = __builtin_amdgcn_wmma_f32_16x16x4_f32(
            false, a, false, b0, (short)0, acc[mt][0], false, false);
        acc[mt][1] = __builtin_amdgcn_wmma_f32_16x16x4_f32(
            false, a, false, b1, (short)0, acc[mt][1], false, false);
      }
    }
    __syncthreads();  // buf free for the DMA issued at iteration c+1
  }

  // Epilogue: sigmoid + store. C layout: VGPR v -> M = v (+8 for lanes 16-31)
  const int hOff = (lane >> 4) * 8;
#pragma unroll
  for (int mt = 0; mt < 8; ++mt)
#pragma unroll
    for (int nt = 0; nt < 2; ++nt)
#pragma unroll
      for (int v = 0; v < 8; ++v) {
        const int token  = tok0 + mt * 16 + v + hOff;
        const int expert = wave * 32 + nt * 16 + l15;
        const float x = acc[mt][nt][v];
        scores[(size_t)token * kExperts + expert] = 1.0f / (1.0f + __expf(-x));
      }
}

// ---------------------------------------------------------------------------
// Kernel 2: grouped top-k routing, one thread per token.
// Reproduces lax.top_k tie-breaking (lowest index wins on ties).
// Output: [32768*8] indices (as float) then [32768*8] weights.
// ---------------------------------------------------------------------------
__launch_bounds__(256, 1)
__global__ void moe_router_topk(const float* __restrict__ scores,
                                const float* __restrict__ bias,
                                float* __restrict__ out) {
  __shared__ float sb[kExperts];
  if (threadIdx.x < kExperts) sb[threadIdx.x] = bias[threadIdx.x];
  __syncthreads();

  const int tok = blockIdx.x * blockDim.x + threadIdx.x;
  if (tok >= kTokens) return;
  const float* s = scores + (size_t)tok * kExperts;

  // group score = sum of top-2 of (score + bias) within each group of 32
  float gs[kGroups];
#pragma unroll
  for (int g = 0; g < kGroups; ++g) {
    float m1 = -INFINITY, m2 = -INFINITY;
    for (int j = 0; j < kGroupSz; ++j) {
      const float v = s[g * kGroupSz + j] + sb[g * kGroupSz + j];
      if (v > m1) { m2 = m1; m1 = v; }
      else if (v > m2) { m2 = v; }
    }
    gs[g] = m1 + m2;
  }

  // top-4 groups via repeated argmax (strict > keeps lowest index on ties)
  unsigned gm = 0;
  for (int r = 0; r < kTopGroup; ++r) {
    float best = -INFINITY; int bi = 0;
#pragma unroll
    for (int g = 0; g < kGroups; ++g)
      if (!((gm >> g) & 1u) && gs[g] > best) { best = gs[g]; bi = g; }
    gm |= 1u << bi;
  }

  // top-8 experts over masked scores (masked-out -> 0.0, like the reference);
  // stable insertion keeps earlier index first on equal values
  float val[kTopK]; int idx[kTopK];
#pragma unroll
  for (int i = 0; i < kTopK; ++i) { val[i] = -INFINITY; idx[i] = 0; }
  for (int e = 0; e < kExperts; ++e) {
    const float v = ((gm >> (e >> 5)) & 1u) ? (s[e] + sb[e]) : 0.0f;
    if (v > val[kTopK - 1]) {
      val[kTopK - 1] = v; idx[kTopK - 1] = e;
#pragma unroll
      for (int p = kTopK - 1; p > 0; --p) {
        if (val[p] > val[p - 1]) {
          const float tv = val[p]; val[p] = val[p - 1]; val[p - 1] = tv;
          const int   ti = idx[p]; idx[p] = idx[p - 1]; idx[p - 1] = ti;
        }
      }
    }
  }

  // gather raw sigmoid scores at chosen experts, normalize, scale
  float w[kTopK]; float sum = 0.0f;
#pragma unroll
  for (int i = 0; i < kTopK; ++i) { w[i] = s[idx[i]]; sum += w[i]; }
  const float scl = kScale / (sum + 1e-20f);

  float* oi = out;                             // indices (as float)
  float* ow = out + (size_t)kTokens * kTopK;   // weights
#pragma unroll
  for (int i = 0; i < kTopK; ++i) {
    oi[(size_t)tok * kTopK + i] = (float)idx[i];
    ow[(size_t)tok * kTopK + i] = w[i] * scl;
  }
}

// ---------------------------------------------------------------------------
extern "C" void kernel_launch(void* const* d_in, const int* in_sizes, int n_in,
                              void* d_out, int out_size, void* d_ws, size_t ws_size,
                              hipStream_t stream) {
  const float* hidden = (const float*)d_in[0];  // [32768, 4096] fp32
  const float* weight = (const float*)d_in[1];  // [256, 4096]  fp32
  const float* bias   = (const float*)d_in[2];  // [256]        fp32
  float* out = (float*)d_out;                   // [32768*8 idx | 32768*8 w]
  float* scores = (float*)d_ws;                 // needs 32768*256*4 = 32 MiB
  (void)in_sizes; (void)n_in; (void)out_size; (void)ws_size;

  moe_router_gemm<<<kTokens / BT, 256, 0, stream>>>(hidden, weight, scores);
  moe_router_topk<<<kTokens / 256, 256, 0, stream>>>(scores, bias, out);
}